// WhModel_46608985096210
// MI455X (gfx1250) — compile-verified
//
#include <hip/hip_runtime.h>

typedef __attribute__((ext_vector_type(2))) float v2f;
typedef __attribute__((ext_vector_type(8))) float v8f;

#define TT 1000000
#define HD 5
#define NB 4
#define CH 16   // timesteps per FC/WMMA flush (T % CH == 0)

__device__ __forceinline__ float bcast_lane(float v, int srcByteIdx) {
  return __builtin_bit_cast(float,
      __builtin_amdgcn_ds_bpermute(srcByteIdx, __builtin_bit_cast(int, v)));
}
// sigmoid(x) = 1/(1+exp(-x)); exp(-x)->0 or inf are both safe through rcp
__device__ __forceinline__ float sigm(float x) {
  float e = __builtin_amdgcn_exp2f(-1.4426950408889634f * x);
  return __builtin_amdgcn_rcpf(1.0f + e);
}
// tanh via |x| so e in (0,1]: no inf-inf
__device__ __forceinline__ float tanh_f(float x) {
  float e = __builtin_amdgcn_exp2f(-2.8853901631880107f * __builtin_fabsf(x));
  float r = (1.0f - e) * __builtin_amdgcn_rcpf(1.0f + e);
  return __builtin_copysignf(r, x);
}

// CDNA5 async memory->LDS copy (VGLOBAL GLOBAL_LOAD_ASYNC_TO_LDS_B32, ASYNCcnt).
// Low 32 bits of a flat LDS address are the LDS byte offset (ISA 10.2 aperture rules).
__device__ __forceinline__ void async_f32_to_lds(const float* gsrc, float* ldst) {
  unsigned int loff = (unsigned int)(unsigned long long)ldst;
  unsigned long long gaddr = (unsigned long long)gsrc;
  asm volatile("global_load_async_to_lds_b32 %0, %1, off"
               :: "v"(loff), "v"(gaddr) : "memory");
}
__device__ __forceinline__ void wait_async0() {
  asm volatile("s_wait_asynccnt 0x0" ::: "memory");
}

__global__ __launch_bounds__(32)
void lstm2_seq_wmma_kernel(const float* __restrict__ xin,
                           const float* __restrict__ wih0, const float* __restrict__ whh0,
                           const float* __restrict__ bih0, const float* __restrict__ bhh0,
                           const float* __restrict__ wih1, const float* __restrict__ whh1,
                           const float* __restrict__ bih1, const float* __restrict__ bhh1,
                           const float* __restrict__ fc1w, const float* __restrict__ fc1b,
                           const float* __restrict__ fc2w, const float* __restrict__ fc2b,
                           float* __restrict__ out) {
  __shared__ float xbuf[CH * NB];   // staged inputs: 16 steps x 4 batches
  __shared__ float h1s[CH * 32];    // staged layer-1 h: 16 steps x (20 used of 32)

  const int lane = (int)threadIdx.x;            // 0..31, single wave32
  const int ls   = (lane < NB * HD) ? lane : 0; // lanes 20..31 mirror lane 0 (harmless)
  const int bb   = ls / HD;                     // batch 0..3
  const int jj   = ls % HD;                     // hidden index 0..4

  // ---- per-lane weights: 4 gate rows r = gi*5 + jj ----
  float Wi0[4], Bs0[4], Bs1[4];
  float Wh0[4][HD], Wi1[4][HD], Wh1[4][HD];
  #pragma unroll
  for (int gi = 0; gi < 4; ++gi) {
    int r = gi * HD + jj;
    Wi0[gi] = wih0[r];                 // input size of layer0 is 1
    Bs0[gi] = bih0[r] + bhh0[r];
    Bs1[gi] = bih1[r] + bhh1[r];
    #pragma unroll
    for (int k = 0; k < HD; ++k) {
      Wh0[gi][k] = whh0[r * HD + k];
      Wi1[gi][k] = wih1[r * HD + k];
      Wh1[gi][k] = whh1[r * HD + k];
    }
  }

  // ---- fold fc2∘fc1 into one 20-vector (exact; both layers are linear) ----
  float weff[20];
  #pragma unroll
  for (int k = 0; k < 20; ++k) {
    float s = 0.f;
    #pragma unroll
    for (int j2 = 0; j2 < 10; ++j2) s = fmaf(fc2w[j2], fc1w[j2 * 20 + k], s);
    weff[k] = s;
  }
  float beff = fc2b[0];
  #pragma unroll
  for (int j2 = 0; j2 < 10; ++j2) beff = fmaf(fc2w[j2], fc1b[j2], beff);

  // bpermute byte indices for broadcasting h[bb][0..4]
  int srcIdx[HD];
  #pragma unroll
  for (int k = 0; k < HD; ++k) srcIdx[k] = (bb * HD + k) << 2;

  float h0v = 0.f, c0v = 0.f, h1v = 0.f, c1v = 0.f;
  const int mrow  = lane & 15;  // WMMA A-matrix M row (timestep in chunk)
  const int khalf = lane >> 4;  // 0: K%4 in {0,1}; 1: K%4 in {2,3}

  for (int t0 = 0; t0 < TT; t0 += CH) {
    // stage 16 steps of input (64 floats) via async global->LDS DMA
    async_f32_to_lds(&xin[t0 * NB + lane],      &xbuf[lane]);
    async_f32_to_lds(&xin[t0 * NB + 32 + lane], &xbuf[32 + lane]);
    if (t0 + CH < TT) __builtin_prefetch(&xin[(t0 + CH) * NB + lane], 0, 1);
    wait_async0();

    #pragma unroll 1
    for (int tt = 0; tt < CH; ++tt) {
      float xv = xbuf[tt * NB + bb];

      // ---------- layer 0 ----------
      float hk[HD];
      #pragma unroll
      for (int k = 0; k < HD; ++k) hk[k] = bcast_lane(h0v, srcIdx[k]);
      float g[4];
      #pragma unroll
      for (int gi = 0; gi < 4; ++gi) {
        float a = fmaf(Wi0[gi], xv, Bs0[gi]);
        #pragma unroll
        for (int k = 0; k < HD; ++k) a = fmaf(hk[k], Wh0[gi][k], a);
        g[gi] = a;
      }
      float iv = sigm(g[0]), fv = sigm(g[1]), gv = tanh_f(g[2]), ov = sigm(g[3]);
      c0v = fmaf(fv, c0v, iv * gv);
      h0v = ov * tanh_f(c0v);

      // ---------- layer 1 ----------
      float h0k[HD], h1k[HD];
      #pragma unroll
      for (int k = 0; k < HD; ++k) h0k[k] = bcast_lane(h0v, srcIdx[k]);
      #pragma unroll
      for (int k = 0; k < HD; ++k) h1k[k] = bcast_lane(h1v, srcIdx[k]);
      #pragma unroll
      for (int gi = 0; gi < 4; ++gi) {
        float a = Bs1[gi];
        #pragma unroll
        for (int k = 0; k < HD; ++k) a = fmaf(h0k[k], Wi1[gi][k], a);
        #pragma unroll
        for (int k = 0; k < HD; ++k) a = fmaf(h1k[k], Wh1[gi][k], a);
        g[gi] = a;
      }
      iv = sigm(g[0]); fv = sigm(g[1]); gv = tanh_f(g[2]); ov = sigm(g[3]);
      c1v = fmaf(fv, c1v, iv * gv);
      h1v = ov * tanh_f(c1v);

      h1s[tt * 32 + lane] = h1v;   // lanes 20..31 write unused columns
    }

    // ---------- FC head: Out(16x1) = H1(16x20) @ weff via V_WMMA_F32_16X16X4_F32 ----------
    // A (16x4 f32): lane m<16 holds (M=m, K=4kb+{0,1}); lane 16+m holds (M=m, K=4kb+{2,3})
    // B (4x16 f32): only column N=0 nonzero = weff[4kb+row]
    v8f acc = {};
    #pragma unroll
    for (int kb = 0; kb < 5; ++kb) {
      const int kbase = kb * 4 + khalf * 2;
      v2f a, b;
      a.x = h1s[mrow * 32 + kbase];
      a.y = h1s[mrow * 32 + kbase + 1];
      b.x = (lane == 0) ? weff[kb * 4 + 0] : ((lane == 16) ? weff[kb * 4 + 2] : 0.f);
      b.y = (lane == 0) ? weff[kb * 4 + 1] : ((lane == 16) ? weff[kb * 4 + 3] : 0.f);
      acc = __builtin_amdgcn_wmma_f32_16x16x4_f32(false, a, false, b,
                                                  (short)0, acc, false, false);
    }
    // D column N=0: lane 0 holds M=0..7 in acc[0..7]; lane 16 holds M=8..15.
    // Each writes 8 contiguous floats -> two global_store_b128.
    if ((lane & 15) == 0) {
      float4 lo, hi;
      lo.x = acc[0] + beff; lo.y = acc[1] + beff;
      lo.z = acc[2] + beff; lo.w = acc[3] + beff;
      hi.x = acc[4] + beff; hi.y = acc[5] + beff;
      hi.z = acc[6] + beff; hi.w = acc[7] + beff;
      float4* p = reinterpret_cast<float4*>(out + t0 + khalf * 8);
      p[0] = lo;
      p[1] = hi;
    }
  }
}

extern "C" void kernel_launch(void* const* d_in, const int* in_sizes, int n_in,
                              void* d_out, int out_size, void* d_ws, size_t ws_size,
                              hipStream_t stream) {
  (void)in_sizes; (void)n_in; (void)out_size; (void)d_ws; (void)ws_size;
  lstm2_seq_wmma_kernel<<<1, 32, 0, stream>>>(
      (const float*)d_in[0],  (const float*)d_in[1],  (const float*)d_in[2],
      (const float*)d_in[3],  (const float*)d_in[4],  (const float*)d_in[5],
      (const float*)d_in[6],  (const float*)d_in[7],  (const float*)d_in[8],
      (const float*)d_in[9],  (const float*)d_in[10], (const float*)d_in[11],
      (const float*)d_in[12], (float*)d_out);
}